// DecoupledTextDecoder_50809463112377
// MI455X (gfx1250) — compile-verified
//
#include <hip/hip_runtime.h>
#include <hip/hip_bf16.h>

// ---------------------------------------------------------------------------
// DecoupledTextDecoder on MI455X (gfx1250), wave32 + WMMA f32 16x16x4 + TDM.
// Round-5: staging reads now index the __shared__ array directly so they
// lower to ds_load (round 2-4 went through a generic-pointer array -> flat).
// nB=256, nC=512, HW=512, nT=40, nCls=97, D=nCls+1=98, nRows=nB*nT=10240.
// ---------------------------------------------------------------------------

#define NB     256
#define NC     512
#define HW     512
#define NT     40
#define NCLS   97
#define OUTD   98
#define NROWS  (NB * NT)        // 10240
#define LSTR   516              // LDS row stride (floats), A-side tiles: 516 % 64 == 4
#define CHUNK  32               // staged K-chunk width (floats)
#define WST    36               // staging row stride (floats), mult of 4, conflict-free
#define EPSQ   0.0009f

typedef float v2f __attribute__((ext_vector_type(2)));
typedef float v8f __attribute__((ext_vector_type(8)));
typedef unsigned int v4u __attribute__((ext_vector_type(4)));
typedef int v4i __attribute__((ext_vector_type(4)));
typedef int v8i __attribute__((ext_vector_type(8)));

// Waits via raw SOPP asm: certain mnemonics + act as compiler memory barriers.
#define WAIT_TENSORCNT_1() asm volatile("s_wait_tensorcnt 0x1" ::: "memory")
#define WAIT_TENSORCNT_0() asm volatile("s_wait_tensorcnt 0x0" ::: "memory")
#define WAIT_DSCNT_0()     asm volatile("s_wait_dscnt 0x0" ::: "memory")

// ---- WMMA f32 16x16x4: D = A(16x4) * B(4x16) + C -------------------------
__device__ __forceinline__ v8f wmma_f32(v2f a, v2f b, v8f c) {
    return __builtin_amdgcn_wmma_f32_16x16x4_f32(false, a, false, b, (short)0, c,
                                                 false, false);
}

// A fragment: lane L holds row m=L&15, k = k0 + (L>>4)*2 .. +1  (ds_load_b64)
__device__ __forceinline__ v2f frag_a(const float* S, int m0, int k, int lane) {
    const float* p = S + (m0 + (lane & 15)) * LSTR + k + ((lane >> 4) << 1);
    return *(const v2f*)p;
}
// B fragment: lane L holds col n=L&15, k = k0 + (L>>4)*2 .. +1  (ds_load_b64)
__device__ __forceinline__ v2f frag_b(const float* Sw, int k, int lane) {
    const float* p = Sw + (lane & 15) * WST + k + ((lane >> 4) << 1);
    return *(const v2f*)p;
}

// ---------------------------------------------------------------------------
// TDM: DMA a 16 x CHUNK f32 tile (row stride 512 elems) from global into LDS.
// Descriptor pads +4 dwords every 32 dwords -> LDS row stride WST=36 floats.
// Rows >= rows_valid return zero via tensor_dim1 OOB (protos tail).
// ---------------------------------------------------------------------------
__device__ __forceinline__ void tdm_stage(const float* gsrc, unsigned lds_byte,
                                          int rows_valid) {
    unsigned long long ga = (unsigned long long)(uintptr_t)gsrc;
    v4u g0;
    g0[0] = 1u;                                         // count=1, user mode
    g0[1] = lds_byte;                                   // LDS dest (bytes)
    g0[2] = (unsigned)ga;                               // global_addr[31:0]
    g0[3] = (unsigned)((ga >> 32) & 0x1FFFFFFu) | (2u << 30);  // addr hi | type=2
    v8i g1;
    g1[0] = (2 << 16)      // data_size = 4B
          | (1 << 20)      // pad_enable
          | (4 << 22)      // pad_interval: every 32 dwords
          | (3 << 25);     // pad_amount: 4 dwords  -> stride 36 floats
    g1[1] = (int)(512u << 16);                          // tensor_dim0 = 512 (lo16)
    g1[2] = (int)((unsigned)rows_valid << 16);          // dim0 hi=0 | tensor_dim1 lo16
    g1[3] = (int)((unsigned)CHUNK << 16);               // dim1 hi=0 | tile_dim0 = 32
    g1[4] = 16;                                         // tile_dim1 = 16, tile_dim2 = 0
    g1[5] = HW;                                         // tensor_dim0_stride = 512
    g1[6] = 0;
    g1[7] = 0;
    v4i g2 = {1, 0, 0, 0};                              // tensor_dim2=1 (unused)
    v4i g3 = {0, 0, 0, 0};
#if defined(__clang_major__) && (__clang_major__ >= 23)
    v8i g4 = {0, 0, 0, 0, 0, 0, 0, 0};
    __builtin_amdgcn_tensor_load_to_lds(g0, g1, g2, g3, g4, 0);
#else
    __builtin_amdgcn_tensor_load_to_lds(g0, g1, g2, g3, 0);
#endif
}

// ---------------------------------------------------------------------------
// K0: zero-fill packed output (harness poisons d_out with 0xAA).
// ---------------------------------------------------------------------------
__global__ void zero_kernel_v5(float* __restrict__ out, int n) {
    int i = blockIdx.x * blockDim.x + threadIdx.x;
    if (i < n) out[i] = 0.0f;
}

// ---------------------------------------------------------------------------
// K1: invSum[b*NT+t] = 1 / sum_hw A[b,t,hw]     (one wave32 per row)
// ---------------------------------------------------------------------------
__global__ __launch_bounds__(256) void rowsum_kernel_v5(const float* __restrict__ A,
                                                        float* __restrict__ invSum) {
    int row  = blockIdx.x * 8 + (threadIdx.x >> 5);
    int lane = threadIdx.x & 31;
    const float* p = A + (size_t)row * HW;
    float s = 0.f;
#pragma unroll
    for (int j = 0; j < HW / 32; ++j) s += p[lane + 32 * j];
#pragma unroll
    for (int off = 16; off > 0; off >>= 1) s += __shfl_xor(s, off, 32);
    if (lane == 0) invSum[row] = 1.0f / s;
}

// ---------------------------------------------------------------------------
// K2: usedLen / exclusive-prefix offsets over textLength (tiny, serial).
// ---------------------------------------------------------------------------
__global__ void offsets_kernel_v5(const int* __restrict__ textLength,
                                  int* __restrict__ offsets,
                                  int* __restrict__ usedLen) {
    if (threadIdx.x == 0 && blockIdx.x == 0) {
        int acc = 0;
        for (int b = 0; b < NB; ++b) {
            int u = textLength[b];
            u = (u < 0) ? 0 : (u > NT ? NT : u);
            usedLen[b] = u;
            offsets[b] = acc;
            acc += u;
        }
    }
}

// ---------------------------------------------------------------------------
// K3: attention pooling as batched GEMM.
//   block = (batch b, N-half). Normalized A[b] (48x512, rows>=40 zero) held in
//   LDS; F streamed once via double-buffered TDM tiles; 3 M-tiles share each
//   B fragment.  C[b*NT+t][c] written for t < NT.
// ---------------------------------------------------------------------------
__global__ __launch_bounds__(256) void pool_gemm_kernel_v5(
        const float* __restrict__ F, const float* __restrict__ A,
        const float* __restrict__ invSum, float* __restrict__ C) {
    __shared__ float sA[48 * LSTR];              // ~99 KB
    __shared__ float sF[8][2][16 * WST];         // ~36.9 KB double-buffered staging

    int b      = blockIdx.x >> 1;
    int cbase0 = (blockIdx.x & 1) * 256;
    int tid    = threadIdx.x;
    int wave   = __builtin_amdgcn_readfirstlane(tid >> 5);   // scalar for TDM desc
    int lane   = tid & 31;

    // Load + normalize the full per-batch attention tile (rows 40..47 = 0).
    for (int idx = tid; idx < 48 * HW; idx += 256) {
        int t = idx >> 9, k = idx & (HW - 1);
        float v = 0.f;
        if (t < NT) v = A[((size_t)b * NT + t) * HW + k] * invSum[b * NT + t];
        sA[t * LSTR + k] = v;
    }
    __syncthreads();

    unsigned lb[2] = { (unsigned)(uintptr_t)&sF[wave][0][0],
                       (unsigned)(uintptr_t)&sF[wave][1][0] };

    for (int nt = wave; nt < 16; nt += 8) {
        int c0 = cbase0 + nt * 16;
        const float* Fbase = F + ((size_t)b * NC + c0) * HW;
        v8f acc0 = {}, acc1 = {}, acc2 = {};

        tdm_stage(Fbase, lb[0], 16);                         // preload chunk 0
        for (int kc = 0; kc < HW; kc += CHUNK) {
            int cur = (kc / CHUNK) & 1;
            if (kc + CHUNK < HW) {
                WAIT_DSCNT_0();                              // done reading buf[cur^1]
                tdm_stage(Fbase + kc + CHUNK, lb[cur ^ 1], 16);
                WAIT_TENSORCNT_1();                          // chunk kc landed
            } else {
                WAIT_TENSORCNT_0();
            }
            const float* Sw = &sF[wave][cur][0];             // LDS-rooted GEP -> ds_load
#pragma unroll
            for (int k0 = 0; k0 < CHUNK; k0 += 4) {
                v2f bf = frag_b(Sw, k0, lane);
                v2f a0 = frag_a(sA, 0,  kc + k0, lane);
                v2f a1 = frag_a(sA, 16, kc + k0, lane);
                v2f a2 = frag_a(sA, 32, kc + k0, lane);
                acc0 = wmma_f32(a0, bf, acc0);
                acc1 = wmma_f32(a1, bf, acc1);
                acc2 = wmma_f32(a2, bf, acc2);
            }
        }
        // C/D layout: lane L, vgpr v -> (m = v + (L>>4)*8, n = L&15)
        int n = lane & 15, half = lane >> 4;
        int col = c0 + n;
#pragma unroll
        for (int v = 0; v < 8; ++v) {
            int m = v + half * 8;
            C[((size_t)b * NT + m) * NC + col]      = acc0[v];
            C[((size_t)b * NT + 16 + m) * NC + col] = acc1[v];
            int t2 = 32 + m;
            if (t2 < NT) C[((size_t)b * NT + t2) * NC + col] = acc2[v];
        }
    }
}

// ---------------------------------------------------------------------------
// K4: fused head per 16-row tile of C:
//   hidden = C*W^T + b  (kept in LDS) -> row norms -> cf = hidden*protos^T
//   -> scatter packed outCls (= cf*alpha) and outCos (= cf/(|h|+eps)) + UNK.
// ---------------------------------------------------------------------------
__global__ __launch_bounds__(256) void head_kernel_v5(
        const float* __restrict__ C, const float* __restrict__ W,
        const float* __restrict__ bias, const float* __restrict__ protos,
        const float* __restrict__ ALPHA, const float* __restrict__ UNK,
        const int* __restrict__ usedLen, const int* __restrict__ offsets,
        float* __restrict__ outCls, float* __restrict__ outCos) {
    __shared__ float sC[16 * LSTR];
    __shared__ float sH[16 * LSTR];
    __shared__ float sS[8][2][16 * WST];
    __shared__ float sInv[16];
    __shared__ int   sDest[16];

    int row0 = blockIdx.x * 16;
    int tid  = threadIdx.x;
    int wave = __builtin_amdgcn_readfirstlane(tid >> 5);
    int lane = tid & 31;

    for (int idx = tid; idx < 16 * (NC / 4); idx += 256) {
        int r = idx >> 7, q = idx & 127;
        *(float4*)&sC[r * LSTR + q * 4] =
            ((const float4*)(C + (size_t)(row0 + r) * NC))[q];
    }
    if (tid < 16) {
        int i = row0 + tid;
        int b = i / NT, t = i % NT;
        sDest[tid] = (t < usedLen[b]) ? (offsets[b] + t) : -1;
    }
    __syncthreads();

    unsigned lb[2] = { (unsigned)(uintptr_t)&sS[wave][0][0],
                       (unsigned)(uintptr_t)&sS[wave][1][0] };

    // hidden: wave handles 4 of the 32 N-tiles
    for (int nt = wave; nt < NC / 16; nt += 8) {
        int n0 = nt * 16;
        const float* Wbase = W + (size_t)n0 * NC;
        v8f acc = {};
        tdm_stage(Wbase, lb[0], 16);
        for (int kc = 0; kc < NC; kc += CHUNK) {
            int cur = (kc / CHUNK) & 1;
            if (kc + CHUNK < NC) {
                WAIT_DSCNT_0();
                tdm_stage(Wbase + kc + CHUNK, lb[cur ^ 1], 16);
                WAIT_TENSORCNT_1();
            } else {
                WAIT_TENSORCNT_0();
            }
            const float* Sw = &sS[wave][cur][0];
#pragma unroll
            for (int k0 = 0; k0 < CHUNK; k0 += 4) {
                v2f bf = frag_b(Sw, k0, lane);
                v2f af = frag_a(sC, 0, kc + k0, lane);
                acc = wmma_f32(af, bf, acc);
            }
        }
        int n = lane & 15, half = lane >> 4;
        float bb = bias[n0 + n];
#pragma unroll
        for (int v = 0; v < 8; ++v)
            sH[(v + half * 8) * LSTR + n0 + n] = acc[v] + bb;
    }
    __syncthreads();

    if (tid < 16) {
        float s = 0.f;
        for (int k = 0; k < NC; ++k) { float h = sH[tid * LSTR + k]; s += h * h; }
        sInv[tid] = 1.0f / (sqrtf(s) + EPSQ);
    }
    __syncthreads();

    float alpha = ALPHA[0];
    float unk   = UNK[0];

    if (wave < 7) {                          // 7 j-tiles cover 112 >= 97 classes
        int j0 = wave * 16;
        int rv = NCLS - j0; rv = (rv > 16) ? 16 : rv;   // protos rows beyond -> 0
        const float* Pbase = protos + (size_t)j0 * NC;
        v8f acc = {};
        tdm_stage(Pbase, lb[0], rv);
        for (int kc = 0; kc < NC; kc += CHUNK) {
            int cur = (kc / CHUNK) & 1;
            if (kc + CHUNK < NC) {
                WAIT_DSCNT_0();
                tdm_stage(Pbase + kc + CHUNK, lb[cur ^ 1], rv);
                WAIT_TENSORCNT_1();
            } else {
                WAIT_TENSORCNT_0();
            }
            const float* Sw = &sS[wave][cur][0];
#pragma unroll
            for (int k0 = 0; k0 < CHUNK; k0 += 4) {
                v2f bf = frag_b(Sw, k0, lane);
                v2f af = frag_a(sH, 0, kc + k0, lane);
                acc = wmma_f32(af, bf, acc);
            }
        }
        int n = lane & 15, half = lane >> 4;
        int cls = j0 + n;
#pragma unroll
        for (int v = 0; v < 8; ++v) {
            int m = v + half * 8;
            int d = sDest[m];
            if (d >= 0 && cls < NCLS) {
                float cf = acc[v];
                outCls[(size_t)d * OUTD + cls] = cf * alpha;
                outCos[(size_t)d * OUTD + cls] = cf * sInv[m];
            }
        }
    } else {                                 // wave 7: UNK column
        if (lane < 16) {
            int d = sDest[lane];
            if (d >= 0) {
                outCls[(size_t)d * OUTD + NCLS] = unk;
                outCos[(size_t)d * OUTD + NCLS] = unk;
            }
        }
    }
}

// ---------------------------------------------------------------------------
extern "C" void kernel_launch(void* const* d_in, const int* in_sizes, int n_in,
                              void* d_out, int out_size, void* d_ws, size_t ws_size,
                              hipStream_t stream) {
    const float* feature    = (const float*)d_in[0];
    const float* A          = (const float*)d_in[1];
    const float* protos     = (const float*)d_in[2];
    const float* W          = (const float*)d_in[3];
    const float* bias       = (const float*)d_in[4];
    const float* ALPHA      = (const float*)d_in[5];
    const float* UNK        = (const float*)d_in[6];
    const int*   textLength = (const int*)d_in[7];

    float* out    = (float*)d_out;
    float* outCls = out;
    float* outCos = out + (size_t)NROWS * OUTD;

    // Workspace layout: invSum[10240] | offsets[256] | usedLen[256] | C[10240*512]
    float* invSum  = (float*)d_ws;
    int*   offsets = (int*)((char*)d_ws + NROWS * sizeof(float));
    int*   usedLen = offsets + NB;
    float* C       = (float*)((char*)d_ws + 65536);

    int nOut = 2 * NROWS * OUTD;
    zero_kernel_v5<<<(nOut + 255) / 256, 256, 0, stream>>>(out, nOut);
    rowsum_kernel_v5<<<NROWS / 8, 256, 0, stream>>>(A, invSum);
    offsets_kernel_v5<<<1, 64, 0, stream>>>(textLength, offsets, usedLen);
    pool_gemm_kernel_v5<<<NB * 2, 256, 0, stream>>>(feature, A, invSum, C);
    head_kernel_v5<<<NROWS / 16, 256, 0, stream>>>(C, W, bias, protos, ALPHA, UNK,
                                                   usedLen, offsets, outCls, outCos);
}